// MaskPooling_47038481826288
// MI455X (gfx1250) — compile-verified
//
#include <hip/hip_runtime.h>

// Problem constants (from reference: x [1,1024,24,24], mask [1,1,336,336])
#define C_DIM   1024
#define G_IN    24
#define G_OUT   336
#define N_PIX   (G_OUT * G_OUT)   // 112896 pixels (= 441 * 256 exactly)
#define NG      (G_IN * G_IN)     // 576 grid nodes
#define KPAD    16                // clusters padded to one WMMA tile
#define NUM_ITERS 5

typedef float v2f __attribute__((ext_vector_type(2)));
typedef float v8f __attribute__((ext_vector_type(8)));

// ---------------------------------------------------------------------------
// Bilinear interpolation params (half-pixel centers, edge clamp == jax linear
// resize / torch align_corners=False for upsampling).
// ---------------------------------------------------------------------------
__device__ __forceinline__ void interp_params(int p, float& w00, float& w01,
                                              float& w10, float& w11,
                                              int& n00, int& n01, int& n10, int& n11) {
  const int hy = p / G_OUT;
  const int hx = p - hy * G_OUT;
  const float scale = (float)G_IN / (float)G_OUT;   // 1/14
  const float sy = (hy + 0.5f) * scale - 0.5f;
  const float sx = (hx + 0.5f) * scale - 0.5f;
  const float y0f = floorf(sy), x0f = floorf(sx);
  const float fy = sy - y0f,   fx = sx - x0f;
  int y0 = (int)y0f, x0 = (int)x0f;
  const int y0c = min(G_IN - 1, max(0, y0));
  const int y1c = min(G_IN - 1, max(0, y0 + 1));
  const int x0c = min(G_IN - 1, max(0, x0));
  const int x1c = min(G_IN - 1, max(0, x0 + 1));
  w00 = (1.f - fy) * (1.f - fx);
  w01 = (1.f - fy) * fx;
  w10 = fy * (1.f - fx);
  w11 = fy * fx;
  n00 = y0c * G_IN + x0c;  n01 = y0c * G_IN + x1c;
  n10 = y1c * G_IN + x0c;  n11 = y1c * G_IN + x1c;
}

// ---------------------------------------------------------------------------
// 1) Transpose x [C,24,24] -> Xt [576][1024] (row = grid node, coalesced rows)
// ---------------------------------------------------------------------------
__global__ void k_transpose(const float* __restrict__ x, float* __restrict__ Xt) {
  int gid = blockIdx.x * blockDim.x + threadIdx.x;
  if (gid >= NG * C_DIM) return;
  int n = gid >> 10;           // / C_DIM
  int c = gid & (C_DIM - 1);
  Xt[gid] = x[c * NG + n];
}

// ---------------------------------------------------------------------------
// 2) tok_sq[p] = ||token_p||^2 (one wave per pixel); store -1 for invalid
//    (invalid == mask<=0 or all-zero token; such tokens have weight 0).
// ---------------------------------------------------------------------------
__global__ void k_toksq(const float* __restrict__ Xt, const float* __restrict__ mask,
                        float* __restrict__ tokSq) {
  const int wave = threadIdx.x >> 5;
  const int lane = threadIdx.x & 31;
  const int p = blockIdx.x * (blockDim.x >> 5) + wave;
  if (p >= N_PIX) return;
  float w00, w01, w10, w11; int n00, n01, n10, n11;
  interp_params(p, w00, w01, w10, w11, n00, n01, n10, n11);
  const float* r00 = Xt + n00 * C_DIM;
  const float* r01 = Xt + n01 * C_DIM;
  const float* r10 = Xt + n10 * C_DIM;
  const float* r11 = Xt + n11 * C_DIM;
  float acc = 0.f;
  for (int c = lane; c < C_DIM; c += 32) {
    float v = w00 * r00[c] + w01 * r01[c] + w10 * r10[c] + w11 * r11[c];
    acc = fmaf(v, v, acc);
  }
  for (int o = 16; o > 0; o >>= 1) acc += __shfl_xor(acc, o, 32);
  if (lane == 0) {
    bool valid = (mask[p] > 0.0f) && (acc > 0.0f);
    tokSq[p] = valid ? acc : -1.0f;
  }
}

// ---------------------------------------------------------------------------
// 3) First k valid pixel indices in row-major order (matches jnp.nonzero)
// ---------------------------------------------------------------------------
__global__ void k_find_init(const float* __restrict__ tokSq, int* __restrict__ initIdx,
                            int k) {
  if (threadIdx.x != 0 || blockIdx.x != 0) return;
  int cnt = 0;
  for (int p = 0; p < N_PIX && cnt < k; ++p)
    if (tokSq[p] > 0.f) initIdx[cnt++] = p;
  while (cnt < k) initIdx[cnt++] = 0;   // fill_value=0 fallback
}

// ---------------------------------------------------------------------------
// 4) centroids0[kk] = token at initIdx[kk]; pad rows [k,16) = 0
// ---------------------------------------------------------------------------
__global__ void k_init_centroids(const float* __restrict__ Xt, const int* __restrict__ initIdx,
                                 float* __restrict__ Cbuf, int k) {
  int kk = blockIdx.x;
  if (kk >= KPAD) return;
  if (kk >= k) {
    for (int c = threadIdx.x; c < C_DIM; c += blockDim.x) Cbuf[kk * C_DIM + c] = 0.f;
    return;
  }
  int p = initIdx[kk];
  float w00, w01, w10, w11; int n00, n01, n10, n11;
  interp_params(p, w00, w01, w10, w11, n00, n01, n10, n11);
  for (int c = threadIdx.x; c < C_DIM; c += blockDim.x)
    Cbuf[kk * C_DIM + c] = w00 * Xt[n00 * C_DIM + c] + w01 * Xt[n01 * C_DIM + c]
                         + w10 * Xt[n10 * C_DIM + c] + w11 * Xt[n11 * C_DIM + c];
}

// ---------------------------------------------------------------------------
// 5a) csq[kk] = ||C[kk]||^2
// ---------------------------------------------------------------------------
__global__ void k_csq(const float* __restrict__ Cbuf, float* __restrict__ csq) {
  __shared__ float s[256];
  const int kk = blockIdx.x;
  float acc = 0.f;
  for (int c = threadIdx.x; c < C_DIM; c += 256) {
    float v = Cbuf[kk * C_DIM + c];
    acc = fmaf(v, v, acc);
  }
  s[threadIdx.x] = acc;
  __syncthreads();
  for (int o = 128; o > 0; o >>= 1) {
    if (threadIdx.x < o) s[threadIdx.x] += s[threadIdx.x + o];
    __syncthreads();
  }
  if (threadIdx.x == 0) csq[kk] = s[0];
}

// ---------------------------------------------------------------------------
// 5b) G = X @ C^T : [576x1024] x [1024x16] -> [576x16], f32 WMMA 16x16x4.
//     One wave per 16-row tile of G; EXEC all-ones; K-loop of 4.
//     A layout (ISA 7.12.2, 32-bit 16x4): lane = m + 16*(k>>1), vgpr = k&1.
//     D layout: vgpr v -> rows v (lanes 0-15) and v+8 (lanes 16-31), col = lane&15.
// ---------------------------------------------------------------------------
__global__ void k_gemm_G(const float* __restrict__ Xt, const float* __restrict__ Cbuf,
                         float* __restrict__ G) {
  const int lane = threadIdx.x & 31;
  const int m0   = blockIdx.x * 16;
  const int half = lane >> 4;       // 0 or 1
  const int nk   = lane & 15;
  v8f acc = {0.f, 0.f, 0.f, 0.f, 0.f, 0.f, 0.f, 0.f};
#if defined(__gfx1250__) && __has_builtin(__builtin_amdgcn_wmma_f32_16x16x4_f32)
  const int mrow = m0 + nk;
  for (int kb = 0; kb < C_DIM; kb += 4) {
    const int col = kb + 2 * half;
    v2f a = { Xt[mrow * C_DIM + col],  Xt[mrow * C_DIM + col + 1] };   // A[m][k]
    v2f b = { Cbuf[nk * C_DIM + col],  Cbuf[nk * C_DIM + col + 1] };   // B[k][n] = C^T
    acc = __builtin_amdgcn_wmma_f32_16x16x4_f32(false, a, false, b,
                                                (short)0, acc, false, false);
  }
#else
  for (int v = 0; v < 8; ++v) {
    const int m = m0 + v + 8 * half;
    float s = 0.f;
    for (int c = 0; c < C_DIM; ++c) s = fmaf(Xt[m * C_DIM + c], Cbuf[nk * C_DIM + c], s);
    acc[v] = s;
  }
#endif
  for (int v = 0; v < 8; ++v)
    G[(m0 + v + 8 * half) * KPAD + nk] = acc[v];
}

// ---------------------------------------------------------------------------
// 5c) zero A2 (node-weight accumulators) and cnt
// ---------------------------------------------------------------------------
__global__ void k_zero(float* __restrict__ A2, float* __restrict__ cnt) {
  int gid = blockIdx.x * blockDim.x + threadIdx.x;
  if (gid < NG * KPAD) A2[gid] = 0.f;
  if (gid < KPAD) cnt[gid] = 0.f;
}

// ---------------------------------------------------------------------------
// 5d) assignment: d2[p,kk] = tokSq - 2*(W·G)[p,kk] + csq[kk]; argmin; then
//     block-local accumulation of bilinear weights in LDS (ds_add_f32) and a
//     sparse flush to global via native f32 atomics (global_atomic_add_f32).
//     A 256-pixel block touches <= 3 source rows, so the LDS tile is sparse.
// ---------------------------------------------------------------------------
__global__ void __launch_bounds__(256)
k_assign(const float* __restrict__ tokSq, const float* __restrict__ G,
         const float* __restrict__ csq, float* __restrict__ A2,
         float* __restrict__ cnt, int k) {
  __shared__ float sA2[NG * KPAD];   // 36 KB of the WGP's 320 KB LDS
  __shared__ float sCnt[KPAD];
  for (int i = threadIdx.x; i < NG * KPAD; i += 256) sA2[i] = 0.f;
  if (threadIdx.x < KPAD) sCnt[threadIdx.x] = 0.f;
  __syncthreads();

  const int p = blockIdx.x * 256 + threadIdx.x;   // N_PIX == gridDim*256 exactly
  const float ts = (p < N_PIX) ? tokSq[p] : -1.0f;
  if (ts > 0.f) {                                 // weight-0 tokens contribute nothing
    float w00, w01, w10, w11; int n00, n01, n10, n11;
    interp_params(p, w00, w01, w10, w11, n00, n01, n10, n11);
    int best = 0;
    float bestd = 3.402823466e38f;
    for (int kk = 0; kk < k; ++kk) {
      float g = w00 * G[n00 * KPAD + kk] + w01 * G[n01 * KPAD + kk]
              + w10 * G[n10 * KPAD + kk] + w11 * G[n11 * KPAD + kk];
      float d2 = ts - 2.f * g + csq[kk];
      if (d2 < bestd) { bestd = d2; best = kk; }  // strict < == argmin first-wins
    }
    atomicAdd(&sA2[n00 * KPAD + best], w00);      // ds_add_f32 (LDS fp atomic)
    atomicAdd(&sA2[n01 * KPAD + best], w01);
    atomicAdd(&sA2[n10 * KPAD + best], w10);
    atomicAdd(&sA2[n11 * KPAD + best], w11);
    atomicAdd(&sCnt[best], 1.0f);
  }
  __syncthreads();

  if (threadIdx.x < KPAD) {
    float cv = sCnt[threadIdx.x];
    if (cv > 0.f) unsafeAtomicAdd(&cnt[threadIdx.x], cv);
  }
  for (int i = threadIdx.x; i < NG * KPAD; i += 256) {
    float v = sA2[i];
    if (v != 0.f) unsafeAtomicAdd(&A2[i], v);     // global_atomic_add_f32
  }
}

// ---------------------------------------------------------------------------
// 5e) Sums = A^T @ X : [16x576] x [576x1024] -> [16x1024], f32 WMMA 16x16x4.
//     One wave per 16-channel tile; K = 576 nodes.
// ---------------------------------------------------------------------------
__global__ void k_gemm_S(const float* __restrict__ A2, const float* __restrict__ Xt,
                         float* __restrict__ Sums) {
  const int lane = threadIdx.x & 31;
  const int cb   = blockIdx.x * 16;    // channel tile base
  const int half = lane >> 4;
  const int nk   = lane & 15;
  v8f acc = {0.f, 0.f, 0.f, 0.f, 0.f, 0.f, 0.f, 0.f};
#if defined(__gfx1250__) && __has_builtin(__builtin_amdgcn_wmma_f32_16x16x4_f32)
  for (int kb = 0; kb < NG; kb += 4) {
    const int kn = kb + 2 * half;                       // node index for this half
    v2f a = { A2[kn * KPAD + nk],       A2[(kn + 1) * KPAD + nk] };       // A[m][k]=A2[k][m]
    v2f b = { Xt[kn * C_DIM + cb + nk], Xt[(kn + 1) * C_DIM + cb + nk] }; // B[k][n]
    acc = __builtin_amdgcn_wmma_f32_16x16x4_f32(false, a, false, b,
                                                (short)0, acc, false, false);
  }
#else
  for (int v = 0; v < 8; ++v) {
    const int m = v + 8 * half;
    float s = 0.f;
    for (int n = 0; n < NG; ++n) s = fmaf(A2[n * KPAD + m], Xt[n * C_DIM + cb + nk], s);
    acc[v] = s;
  }
#endif
  for (int v = 0; v < 8; ++v)
    Sums[(v + 8 * half) * C_DIM + cb + nk] = acc[v];
}

// ---------------------------------------------------------------------------
// 5f) centroid update: where(cnt>0, sums/max(cnt,1e-30), old)
// ---------------------------------------------------------------------------
__global__ void k_update(const float* __restrict__ Sums, const float* __restrict__ cnt,
                         float* __restrict__ Cbuf, int k) {
  int gid = blockIdx.x * blockDim.x + threadIdx.x;
  if (gid >= KPAD * C_DIM) return;
  int kk = gid >> 10;
  if (kk >= k) return;
  float cv = cnt[kk];
  if (cv > 0.f) Cbuf[gid] = Sums[gid] / fmaxf(cv, 1e-30f);
}

// ---------------------------------------------------------------------------
// 6) emit output
// ---------------------------------------------------------------------------
__global__ void k_output(const float* __restrict__ Cbuf, float* __restrict__ out, int total) {
  int gid = blockIdx.x * blockDim.x + threadIdx.x;
  if (gid < total) out[gid] = Cbuf[gid];
}

extern "C" void kernel_launch(void* const* d_in, const int* in_sizes, int n_in,
                              void* d_out, int out_size, void* d_ws, size_t ws_size,
                              hipStream_t stream) {
  (void)in_sizes; (void)n_in; (void)ws_size;
  const float* x    = (const float*)d_in[0];   // [1,1024,24,24]
  const float* mask = (const float*)d_in[1];   // [1,1,336,336]
  int k = out_size / C_DIM;                    // mask_token_num (10)
  if (k < 1) k = 1;
  if (k > KPAD) k = KPAD;

  // workspace layout (floats), ~3.0 MB total
  float* ws    = (float*)d_ws;
  size_t o = 0;
  float* Xt    = ws + o;  o += (size_t)NG * C_DIM;    // 589824
  float* tokSq = ws + o;  o += (size_t)N_PIX;         // 112896
  float* Cbuf  = ws + o;  o += (size_t)KPAD * C_DIM;  // 16384
  float* csq   = ws + o;  o += KPAD;
  float* Gm    = ws + o;  o += (size_t)NG * KPAD;     // 9216
  float* A2    = ws + o;  o += (size_t)NG * KPAD;     // 9216
  float* cnt   = ws + o;  o += KPAD;
  float* Sums  = ws + o;  o += (size_t)KPAD * C_DIM;  // 16384
  int*   initIdx = (int*)(ws + o);

  k_transpose<<<(NG * C_DIM + 255) / 256, 256, 0, stream>>>(x, Xt);
  k_toksq<<<N_PIX / 8, 256, 0, stream>>>(Xt, mask, tokSq);
  k_find_init<<<1, 32, 0, stream>>>(tokSq, initIdx, k);
  k_init_centroids<<<KPAD, 256, 0, stream>>>(Xt, initIdx, Cbuf, k);

  for (int it = 0; it < NUM_ITERS; ++it) {
    k_csq<<<KPAD, 256, 0, stream>>>(Cbuf, csq);
    k_gemm_G<<<NG / 16, 32, 0, stream>>>(Xt, Cbuf, Gm);
    k_zero<<<(NG * KPAD + 255) / 256, 256, 0, stream>>>(A2, cnt);
    k_assign<<<(N_PIX + 255) / 256, 256, 0, stream>>>(tokSq, Gm, csq, A2, cnt, k);
    k_gemm_S<<<C_DIM / 16, 32, 0, stream>>>(A2, Xt, Sums);
    k_update<<<(KPAD * C_DIM + 255) / 256, 256, 0, stream>>>(Sums, cnt, Cbuf, k);
  }

  k_output<<<(out_size + 255) / 256, 256, 0, stream>>>(Cbuf, (float*)d_out, out_size);
}